// LocalConLoss_35862976922133
// MI455X (gfx1250) — compile-verified
//
#include <hip/hip_runtime.h>
#include <hip/hip_bf16.h>

// ---------------------------------------------------------------------------
// LocalConLoss on MI455X (gfx1250, wave32, WMMA + async-LDS pipeline).
//   N = 8192 rows, C = 64 features, logits = X X^T / 0.7 (8192x8192 gram).
//   Core matmul: V_WMMA_F32_16X16X4_F32 (full fp32, matches reference).
//   Column panels staged to LDS with GLOBAL_LOAD_ASYNC_TO_LDS_B128,
//   double-buffered and paced with s_wait_asynccnt.
// ---------------------------------------------------------------------------

#define NROW   8192
#define CDIM   64
#define MTILE  32               // rows per block (two 16-row WMMA tiles)
#define CHUNK  128              // column rows staged per LDS panel
#define NCHUNK (NROW / CHUNK)   // 64
#define INV_TEMP 1.4285714285714286f   // 1/0.7
#define LOG2E    1.4426950408889634f
#define LN2      0.6931471805599453f

typedef __attribute__((ext_vector_type(2))) float v2f;
typedef __attribute__((ext_vector_type(8))) float v8f;

static __device__ __forceinline__ unsigned lds_off(const void* p) {
    // generic pointers to LDS carry the workgroup-relative byte offset in
    // their low 32 bits; the async-LDS instruction adds the wave's LDS_BASE.
    return (unsigned)(uintptr_t)p;
}

// ---------------------------------------------------------------------------
// Kernel 1: gather strided/transposed X (N x C) and lbl (N) into workspace.
//   X[n][c] = features[b][v][c][4*hh][4*ww],  n = ((v*4+b)*32 + hh)*32 + ww
// ---------------------------------------------------------------------------
__global__ void gather_kernel(const float* __restrict__ features,
                              const int*   __restrict__ labels,
                              float* __restrict__ X,
                              int*   __restrict__ lbl)
{
    int tid = blockIdx.x * blockDim.x + threadIdx.x;   // [0, N*C)
    int n = tid & (NROW - 1);
    int c = tid >> 13;                                 // NROW = 2^13
    int ww = n & 31;
    int hh = (n >> 5) & 31;
    int b  = (n >> 10) & 3;
    int v  = n >> 12;
    // features shape (4,2,64,128,128)
    size_t fidx = ((size_t)((b * 2 + v) * 64 + c)) * 16384u
                + (size_t)(hh * 4) * 128u + (size_t)(ww * 4);
    X[(size_t)n * CDIM + c] = features[fidx];
    if (c == 0) {
        // labels shape (4,2,128,128)
        size_t lidx = (size_t)(b * 2 + v) * 16384u
                    + (size_t)(hh * 4) * 128u + (size_t)(ww * 4);
        lbl[n] = labels[lidx];
    }
}

// ---------------------------------------------------------------------------
// Kernel 2: fused gram-matrix + row statistics.
// Block = 32 rows (two 16-row M-tiles), 8 waves. Columns swept in 64 panels
// of 128 rows staged in double-buffered LDS via async-LDS loads; each wave
// owns one 16-column tile of the panel (j0 = base + wave*16).
//
// fp32 16x16x4 fragment layout (A of rows == B of X^T columns, symmetric):
//   lane L<16 : { X[row+L][4k+0], X[row+L][4k+1] }
//   lane L+16 : { X[row+L][4k+2], X[row+L][4k+3] }
// C/D layout: lane holds column n = lane&15; VGPR r holds row m = r+8*(lane>>4).
// ---------------------------------------------------------------------------
__global__ void __launch_bounds__(256, 1)
conloss_wmma_kernel(const float* __restrict__ X,
                    const int*   __restrict__ lbl,
                    float* __restrict__ bNum,
                    float* __restrict__ bDen)
{
    __shared__ float sbuf[2][CHUNK * CDIM];     // 2 x 32 KB column panels
    __shared__ float sExp[8][MTILE];
    __shared__ float sSL [8][MTILE];
    __shared__ float sCnt[8][MTILE];
    __shared__ float sLoss[MTILE];
    __shared__ float sNb  [MTILE];

    const int i0   = blockIdx.x * MTILE;
    const int wave = threadIdx.x >> 5;
    const int lane = threadIdx.x & 31;
    const int ln16 = lane & 15;
    const int half = lane >> 4;

    // ---- A fragments for the two row tiles (i0..i0+15, i0+16..i0+31) ----
    v2f a0[16], a1[16];
    {
        const float* ap = X + (size_t)(i0 + ln16) * CDIM + half * 2;
#pragma unroll
        for (int k = 0; k < 16; ++k) a0[k] = *(const v2f*)(ap + 4 * k);
        ap += 16 * CDIM;
#pragma unroll
        for (int k = 0; k < 16; ++k) a1[k] = *(const v2f*)(ap + 4 * k);
    }
    // labels of the 16 rows this lane accumulates (8 per M-tile)
    int lblA[16];
#pragma unroll
    for (int r = 0; r < 8; ++r) {
        lblA[r]     = lbl[i0 + half * 8 + r];
        lblA[8 + r] = lbl[i0 + 16 + half * 8 + r];
    }

    float aExp[16], aSL[16], aCnt[16];
#pragma unroll
    for (int r = 0; r < 16; ++r) { aExp[r] = 0.f; aSL[r] = 0.f; aCnt[r] = 0.f; }

    // ---- async panel copy: 32 KB, 8 x b128 per thread ----
    auto issue_copy = [&](int c, int bufi) {
        const float* src = X + (size_t)c * CHUNK * CDIM;
        unsigned ldsbase = lds_off(&sbuf[bufi][0]);
#pragma unroll
        for (int q = 0; q < 8; ++q) {
            const int idx = threadIdx.x + q * 256;     // float4 index
            const float* g = src + (size_t)idx * 4;
            unsigned     l = ldsbase + (unsigned)idx * 16u;
            asm volatile("global_load_async_to_lds_b128 %0, %1, off"
                         :: "v"(l), "v"(g) : "memory");
        }
    };

    issue_copy(0, 0);

    for (int c = 0; c < NCHUNK; ++c) {
        const int cur = c & 1;
        if (c + 1 < NCHUNK) {
            issue_copy(c + 1, cur ^ 1);
            asm volatile("s_wait_asynccnt 0x8" ::: "memory");  // panel c landed
        } else {
            asm volatile("s_wait_asynccnt 0x0" ::: "memory");
        }
        __syncthreads();                                        // panel visible

        const int j0   = c * CHUNK + wave * 16;                 // this wave's tile
        const int lblj = lbl[j0 + ln16];
        const float* bp = &sbuf[cur][(wave * 16 + ln16) * CDIM + half * 2];

        v8f d0 = {};
        v8f d1 = {};
#pragma unroll
        for (int k = 0; k < 16; ++k) {
            const v2f b = *(const v2f*)(bp + 4 * k);            // ds_load_b64
            d0 = __builtin_amdgcn_wmma_f32_16x16x4_f32(
                     false, a0[k], false, b, (short)0, d0, false, false);
            d1 = __builtin_amdgcn_wmma_f32_16x16x4_f32(
                     false, a1[k], false, b, (short)0, d1, false, false);
        }

        // fused epilogue: exp / diagonal exclusion / label mask (branchless)
#pragma unroll
        for (int r = 0; r < 8; ++r) {
            const int j  = j0 + ln16;
            const int i_0 = i0 + half * 8 + r;
            const int i_1 = i_0 + 16;

            const float lg0 = d0[r] * INV_TEMP;
            const bool dg0  = (i_0 == j);
            aExp[r]     += dg0 ? 0.0f : __builtin_amdgcn_exp2f(lg0 * LOG2E);
            const bool sm0 = (!dg0) && (lblA[r] == lblj);
            aSL [r]     += sm0 ? lg0  : 0.0f;
            aCnt[r]     += sm0 ? 1.0f : 0.0f;

            const float lg1 = d1[r] * INV_TEMP;
            const bool dg1  = (i_1 == j);
            aExp[8 + r] += dg1 ? 0.0f : __builtin_amdgcn_exp2f(lg1 * LOG2E);
            const bool sm1 = (!dg1) && (lblA[8 + r] == lblj);
            aSL [8 + r] += sm1 ? lg1  : 0.0f;
            aCnt[8 + r] += sm1 ? 1.0f : 0.0f;
        }
        __syncthreads();   // all waves done reading panel `cur^1`'s successor slot
    }

    // ---- butterfly reduce within each 16-lane half ----
#pragma unroll
    for (int s = 1; s < 16; s <<= 1) {
#pragma unroll
        for (int r = 0; r < 16; ++r) {
            aExp[r] += __shfl_xor(aExp[r], s, 32);
            aSL [r] += __shfl_xor(aSL [r], s, 32);
            aCnt[r] += __shfl_xor(aCnt[r], s, 32);
        }
    }
    if (ln16 == 0) {
#pragma unroll
        for (int r = 0; r < 8; ++r) {
            sExp[wave][half * 8 + r]      = aExp[r];
            sSL [wave][half * 8 + r]      = aSL [r];
            sCnt[wave][half * 8 + r]      = aCnt[r];
            sExp[wave][16 + half * 8 + r] = aExp[8 + r];
            sSL [wave][16 + half * 8 + r] = aSL [8 + r];
            sCnt[wave][16 + half * 8 + r] = aCnt[8 + r];
        }
    }
    __syncthreads();

    if (threadIdx.x < MTILE) {
        const int m = threadIdx.x;
        float eSum = 0.f, lSum = 0.f, cSum = 0.f;
#pragma unroll
        for (int w = 0; w < 8; ++w) {
            eSum += sExp[w][m]; lSum += sSL[w][m]; cSum += sCnt[w][m];
        }
        const int li = lbl[i0 + m];
        float loss = 0.0f;
        if (cSum > 0.0f) {
            const float lg = __builtin_amdgcn_logf(eSum) * LN2;  // ln(rowsum)
            loss = -(lSum - cSum * lg) / cSum;
        }
        const float nb = (li != 0 && li != -1) ? 1.0f : 0.0f;
        sLoss[m] = loss * nb;
        sNb[m]   = nb;
    }
    __syncthreads();

    if (threadIdx.x == 0) {
        float num = 0.f, den = 0.f;
#pragma unroll
        for (int m = 0; m < MTILE; ++m) { num += sLoss[m]; den += sNb[m]; }
        bNum[blockIdx.x] = num;
        bDen[blockIdx.x] = den;
    }
}

// ---------------------------------------------------------------------------
// Kernel 3: deterministic final reduction + labels.sum()==0 guard.
// ---------------------------------------------------------------------------
__global__ void finalize_kernel(const int*   __restrict__ lbl,
                                const float* __restrict__ bNum,
                                const float* __restrict__ bDen,
                                float* __restrict__ out)
{
    __shared__ float sN[256];
    __shared__ float sD[256];
    __shared__ int   sL[256];
    const int t = threadIdx.x;
    float n = 0.f, d = 0.f;
    int ls = 0;
    for (int k = t; k < NROW / MTILE; k += 256) { n += bNum[k]; d += bDen[k]; }
    for (int k = t; k < NROW; k += 256) ls += lbl[k];
    sN[t] = n; sD[t] = d; sL[t] = ls;
    __syncthreads();
    for (int s = 128; s > 0; s >>= 1) {
        if (t < s) { sN[t] += sN[t + s]; sD[t] += sD[t + s]; sL[t] += sL[t + s]; }
        __syncthreads();
    }
    if (t == 0)
        out[0] = (sL[0] == 0) ? 0.0f : sN[0] / sD[0];
}

// ---------------------------------------------------------------------------
extern "C" void kernel_launch(void* const* d_in, const int* in_sizes, int n_in,
                              void* d_out, int out_size, void* d_ws, size_t ws_size,
                              hipStream_t stream)
{
    (void)in_sizes; (void)n_in; (void)out_size; (void)ws_size;
    const float* features = (const float*)d_in[0];
    const int*   labels   = (const int*)d_in[1];
    float*       out      = (float*)d_out;

    char* ws = (char*)d_ws;
    float* X    = (float*)ws;                                   // 2 MB
    int*   lbl  = (int*)(ws + (size_t)NROW * CDIM * 4);         // 32 KB
    float* bNum = (float*)(ws + (size_t)NROW * CDIM * 4 + NROW * 4);
    float* bDen = bNum + (NROW / MTILE);

    gather_kernel<<<(NROW * CDIM) / 256, 256, 0, stream>>>(features, labels, X, lbl);
    conloss_wmma_kernel<<<NROW / MTILE, 256, 0, stream>>>(X, lbl, bNum, bDen);
    finalize_kernel<<<1, 256, 0, stream>>>(lbl, bNum, bDen, out);
}